// LSTMCustom_39943195853068
// MI455X (gfx1250) — compile-verified
//
#include <hip/hip_runtime.h>
#include <math.h>

// ---------------------------------------------------------------------------
// LSTM on MI455X (gfx1250), fp32 end-to-end via V_WMMA_F32_16X16X4_F32.
//
//   T=256, B=64, D=1024, H=1024
//   Phase 1: xproj[(t,b), g] = x · Wx^T   (16384 x 1024 x 4096 GEMM)
//            64x64 register-blocked tiles: 16 WMMA per 8 loads (2 FLOP-dense
//            vs the naive 4 WMMA : 5 loads), since this phase is L2-BW bound.
//   Phase 2: 256 sequential step kernels; each wave computes a 16x16
//            (batch x hidden) tile of all 4 gates with WMMA (C operand
//            seeded from xproj), applies sigmoid/tanh, updates c in place,
//            writes h_t into h_all.  Kept at max parallelism (256 waves).
//   Phase 3: out = h_all · Wout^T         (16384 x 1024 x 1024 GEMM), same
//            64x64 blocking.
//
// Workspace layout (floats):
//   xproj : T*B*4H = 67,108,864      (268 MB)
//   h_all : T*B*H  = 16,777,216      ( 67 MB)
//   c     : B*H    =     65,536      (0.25 MB)
// ---------------------------------------------------------------------------

typedef __attribute__((ext_vector_type(2))) float v2f;
typedef __attribute__((ext_vector_type(8))) float v8f;

#define TT 256
#define BB 64
#define DD 1024
#define HH 1024
#define GG (4 * HH)

__device__ __forceinline__ v8f wmma_k4(v2f a, v2f b, v8f c) {
    // D = A(16x4) * B(4x16) + C(16x16), all fp32, RNE.
    return __builtin_amdgcn_wmma_f32_16x16x4_f32(false, a, false, b, (short)0, c,
                                                 false, false);
}

__device__ __forceinline__ float sigmoid_f(float x) {
    return 1.0f / (1.0f + expf(-x));
}

// ---------------------------------------------------------------------------
// 64x64 register-blocked fp32 GEMM tile for one wave:
//   Cout[0..63, 0..63] = A[0..63, 0..K-1] * W[0..63, 0..K-1]^T
// A is row-major (lda), W row-major (ldb) used as B with B[k,n] = W[n,k].
// 16 accumulators (128 VGPRs); inner step: 8x global_load_b64 -> 16x WMMA.
// ---------------------------------------------------------------------------
__device__ __forceinline__ void gemm_tile64(const float* __restrict__ A, int lda,
                                            const float* __restrict__ W, int ldb,
                                            float* __restrict__ Cout, int ldc,
                                            int K, int lane)
{
    const int mlane = lane & 15;
    const int khalf = (lane >> 4) * 2;
    const int rsel  = (lane >> 4) * 8;

    const float* ar[4];
    const float* br[4];
#pragma unroll
    for (int i = 0; i < 4; ++i) {
        ar[i] = A + (size_t)(i * 16 + mlane) * lda + khalf;
        br[i] = W + (size_t)(i * 16 + mlane) * ldb + khalf;
    }

    v8f acc[16];
#pragma unroll
    for (int i = 0; i < 16; ++i) acc[i] = (v8f){};

#pragma unroll 2
    for (int k = 0; k < K; k += 4) {
        v2f a[4], b[4];
#pragma unroll
        for (int i = 0; i < 4; ++i) a[i] = *(const v2f*)(ar[i] + k);
#pragma unroll
        for (int i = 0; i < 4; ++i) b[i] = *(const v2f*)(br[i] + k);
#pragma unroll
        for (int mi = 0; mi < 4; ++mi)
#pragma unroll
            for (int ni = 0; ni < 4; ++ni)
                acc[mi * 4 + ni] = wmma_k4(a[mi], b[ni], acc[mi * 4 + ni]);
    }

#pragma unroll
    for (int mi = 0; mi < 4; ++mi) {
#pragma unroll
        for (int v = 0; v < 8; ++v) {
            float* orow = Cout + (size_t)(mi * 16 + v + rsel) * ldc + mlane;
#pragma unroll
            for (int ni = 0; ni < 4; ++ni)
                orow[ni * 16] = acc[mi * 4 + ni][v];
        }
    }
}

// ---------------------------------------------------------------------------
// Phase 1: xproj GEMM.  One wave -> 64(M) x 64(N) tile.  The 64-wide N group
// never crosses a gate boundary (64 | 1024), so each wave reads exactly one
// of the four weight matrices.
// ---------------------------------------------------------------------------
__global__ void __launch_bounds__(256)
lstm_xproj_kernel(const float* __restrict__ x,
                  const float* __restrict__ Wxi, const float* __restrict__ Wxf,
                  const float* __restrict__ Wxo, const float* __restrict__ Wxc,
                  float* __restrict__ xproj)
{
    const int wave  = (blockIdx.x * blockDim.x + threadIdx.x) >> 5;
    const int lane  = threadIdx.x & 31;
    const int NGRP  = GG / 64;               // 64 column groups
    const int mtile = wave / NGRP;           // 0..255   (64-row blocks of T*B)
    const int ngrp  = wave % NGRP;
    const int row0  = mtile * 64;
    const int gcol0 = ngrp * 64;

    const float* Wq;
    switch (gcol0 / HH) {
        case 0:  Wq = Wxi; break;
        case 1:  Wq = Wxf; break;
        case 2:  Wq = Wxo; break;
        default: Wq = Wxc; break;
    }
    const int col0 = gcol0 % HH;

    gemm_tile64(x + (size_t)row0 * DD, DD,
                Wq + (size_t)col0 * DD, DD,
                xproj + (size_t)row0 * GG + gcol0, GG,
                DD, lane);
}

// ---------------------------------------------------------------------------
// Phase 2: one step of the recurrence.  256 waves total (4 mtiles x 64
// ntiles); each wave computes the i/f/o/c-tilde gate tiles for its 16x16
// (batch x hidden) region, seeded from xproj, then does the pointwise LSTM
// update on its own disjoint region of c / h.
// ---------------------------------------------------------------------------
__global__ void __launch_bounds__(256)
lstm_step_kernel(const float* __restrict__ hprev,    // (B,H)
                 const float* __restrict__ xproj_t,  // (B,4H) slice at t
                 const float* __restrict__ Whi, const float* __restrict__ Whf,
                 const float* __restrict__ Who, const float* __restrict__ Whc,
                 float* __restrict__ cstate,         // (B,H), updated in place
                 float* __restrict__ hout)           // h_all + t*B*H
{
    const int wave  = (blockIdx.x * blockDim.x + threadIdx.x) >> 5;  // 0..255
    const int lane  = threadIdx.x & 31;
    const int NT    = HH / 16;               // 64 hidden tiles
    const int mtile = wave / NT;             // 0..3 (batch tiles)
    const int ntile = wave % NT;
    const int row0  = mtile * 16;
    const int col0  = ntile * 16;

    const int mlane = lane & 15;
    const int khalf = (lane >> 4) * 2;
    const int rsel  = (lane >> 4) * 8;

    // Seed accumulators with the precomputed input projections (C operand).
    v8f ai, af, ao, ag;
#pragma unroll
    for (int v = 0; v < 8; ++v) {
        const float* xr = xproj_t + (size_t)(row0 + v + rsel) * GG + col0 + mlane;
        ai[v] = xr[0 * HH];
        af[v] = xr[1 * HH];
        ao[v] = xr[2 * HH];
        ag[v] = xr[3 * HH];
    }

    const float* arow = hprev + (size_t)(row0 + mlane) * HH + khalf;
    const float* bi   = Whi + (size_t)(col0 + mlane) * HH + khalf;
    const float* bf   = Whf + (size_t)(col0 + mlane) * HH + khalf;
    const float* bo   = Who + (size_t)(col0 + mlane) * HH + khalf;
    const float* bg   = Whc + (size_t)(col0 + mlane) * HH + khalf;

#pragma unroll 8
    for (int k = 0; k < HH; k += 4) {
        v2f a  = *(const v2f*)(arow + k);
        v2f wi = *(const v2f*)(bi + k);
        v2f wf = *(const v2f*)(bf + k);
        v2f wo = *(const v2f*)(bo + k);
        v2f wg = *(const v2f*)(bg + k);
        ai = wmma_k4(a, wi, ai);
        af = wmma_k4(a, wf, af);
        ao = wmma_k4(a, wo, ao);
        ag = wmma_k4(a, wg, ag);
    }

    // Pointwise LSTM cell update on this wave's disjoint 16x16 region.
#pragma unroll
    for (int v = 0; v < 8; ++v) {
        const int r  = row0 + v + rsel;
        const int cc = col0 + mlane;
        const size_t idx = (size_t)r * HH + cc;
        float gi = sigmoid_f(ai[v]);
        float gf = sigmoid_f(af[v]);
        float go = sigmoid_f(ao[v]);
        float gc = tanhf(ag[v]);
        float cn = gf * cstate[idx] + gi * gc;
        cstate[idx] = cn;
        hout[idx]   = go * tanhf(cn);
    }
}

// ---------------------------------------------------------------------------
// Phase 3: output projection over all timesteps at once:
//   out[(t,b), n] = sum_k h_all[(t,b), k] * Wout[n, k]
// ---------------------------------------------------------------------------
__global__ void __launch_bounds__(256)
lstm_out_kernel(const float* __restrict__ h_all,
                const float* __restrict__ Wout,
                float* __restrict__ out)
{
    const int wave  = (blockIdx.x * blockDim.x + threadIdx.x) >> 5;
    const int lane  = threadIdx.x & 31;
    const int NGRP  = HH / 64;               // 16 column groups
    const int mtile = wave / NGRP;           // 0..255
    const int ngrp  = wave % NGRP;
    const int row0  = mtile * 64;
    const int col0  = ngrp * 64;

    gemm_tile64(h_all + (size_t)row0 * HH, HH,
                Wout + (size_t)col0 * HH, HH,
                out + (size_t)row0 * HH + col0, HH,
                HH, lane);
}

// ---------------------------------------------------------------------------
extern "C" void kernel_launch(void* const* d_in, const int* in_sizes, int n_in,
                              void* d_out, int out_size, void* d_ws, size_t ws_size,
                              hipStream_t stream)
{
    const float* x    = (const float*)d_in[0];
    const float* h0   = (const float*)d_in[1];
    const float* c0   = (const float*)d_in[2];
    const float* Wxi  = (const float*)d_in[3];
    const float* Wxf  = (const float*)d_in[4];
    const float* Wxo  = (const float*)d_in[5];
    const float* Wxc  = (const float*)d_in[6];
    const float* Whi  = (const float*)d_in[7];
    const float* Whf  = (const float*)d_in[8];
    const float* Who  = (const float*)d_in[9];
    const float* Whc  = (const float*)d_in[10];
    const float* Wout = (const float*)d_in[11];
    float* out = (float*)d_out;

    float* xproj = (float*)d_ws;                               // T*B*4H
    float* h_all = xproj + (size_t)TT * BB * GG;               // T*B*H
    float* cbuf  = h_all + (size_t)TT * BB * HH;               // B*H

    // c state initialized from c0 each call (deterministic).
    hipMemcpyAsync(cbuf, c0, sizeof(float) * BB * HH,
                   hipMemcpyDeviceToDevice, stream);

    // Phase 1: (16384/64) mtiles * (4096/64) ngroups = 16384 waves -> 2048 blocks.
    lstm_xproj_kernel<<<2048, 256, 0, stream>>>(x, Wxi, Wxf, Wxo, Wxc, xproj);

    // Phase 2: 256 sequential steps; 256 waves each -> 32 blocks of 8 waves.
    for (int t = 0; t < TT; ++t) {
        const float* hprev = (t == 0) ? h0 : (h_all + (size_t)(t - 1) * BB * HH);
        lstm_step_kernel<<<32, 256, 0, stream>>>(
            hprev, xproj + (size_t)t * BB * GG,
            Whi, Whf, Who, Whc,
            cbuf, h_all + (size_t)t * BB * HH);
    }

    // Phase 3: (16384/64) mtiles * (1024/64) ngroups = 4096 waves -> 512 blocks.
    lstm_out_kernel<<<512, 256, 0, stream>>>(h_all, Wout, out);
}